// decoder_80753975099643
// MI455X (gfx1250) — compile-verified
//
#include <hip/hip_runtime.h>
#include <math.h>

// Shapes (fixed by reference): S=16384, B=1, E=2048, H=1024, C=5.
// softmax over a size-1 axis == 1.0, so attention collapses to a column sum
// of encoder_hiddens. Everything is bandwidth-bound f32 (~208 MB total).

typedef __attribute__((ext_vector_type(2))) float v2f;
typedef __attribute__((ext_vector_type(8))) float v8f;

#define E_DIM 2048
#define H_DIM 1024
#define S_DIM 16384
#define NSEG  128
#define ROWS_PER_SEG (S_DIM / NSEG)   // 128
#define GATES 4096

// ---------------------------------------------------------------------------
// Phase 1a: partial column sums of encoder_hiddens (S x E), f32.
// grid (2, NSEG), block 256; each thread does 4 consecutive columns (b128
// loads). Dominant cost of the whole problem: 128 MB HBM stream.
// ---------------------------------------------------------------------------
__global__ void k_col_partial(const float* __restrict__ eh, float* __restrict__ part) {
    int col = blockIdx.x * 1024 + threadIdx.x * 4;
    int seg = blockIdx.y;
    size_t base = (size_t)seg * ROWS_PER_SEG * E_DIM + col;
    float4 acc = make_float4(0.f, 0.f, 0.f, 0.f);
    #pragma unroll 4
    for (int r = 0; r < ROWS_PER_SEG; ++r) {
        // speculative prefetch a few rows ahead (global_prefetch_b8)
        __builtin_prefetch(eh + base + (size_t)(r + 8) * E_DIM, 0, 0);
        float4 v = *(const float4*)(eh + base + (size_t)r * E_DIM);
        acc.x += v.x; acc.y += v.y; acc.z += v.z; acc.w += v.w;
    }
    *(float4*)(part + (size_t)seg * E_DIM + col) = acc;
}

// Phase 1b: combine segment partials -> attn_applied x0 (2048 floats)
__global__ void k_col_reduce(const float* __restrict__ part, float* __restrict__ x0) {
    int col = blockIdx.x * 256 + threadIdx.x;
    float s = 0.f;
    for (int seg = 0; seg < NSEG; ++seg) s += part[(size_t)seg * E_DIM + col];
    x0[col] = s;
}

// ---------------------------------------------------------------------------
// GEMV via V_WMMA_F32_16X16X4_F32.  D(16x16) = A(16x4) * B(4x16) + C.
// A = weight tile (rows = M, documented layout: lane = M%16, VGPR0/1 hold
//     K = {0,1} for lanes 0-15 and K = {2,3} for lanes 16-31).
// B = x vector in column N=0 only: lane0 carries K0/K1, lane16 carries K2/K3,
//     all other lanes zero.  Column 0 of D -> lane0 (M=0..7), lane16 (M=8..15).
// One wave per block; grid = (rows/16, kChunks); partial sums per k-chunk.
// Unrolled x4 with 4 independent accumulators to break the D->C WMMA chain.
// ---------------------------------------------------------------------------
__global__ void k_gemv_wmma(const float* __restrict__ W, const float* __restrict__ x,
                            int Kdim, int kChunk, float* __restrict__ pOut) {
    int lane = threadIdx.x;              // 0..31, one wave32 per block
    int row0 = blockIdx.x * 16;
    int m    = lane & 15;
    int koff = (lane >> 4) << 1;         // 0 for lanes 0-15, 2 for lanes 16-31
    const float* wrow = W + (size_t)(row0 + m) * Kdim + koff;

    int k0 = blockIdx.y * kChunk;
    int k1 = k0 + kChunk;                // kChunk is a multiple of 16

    bool isL0  = (lane == 0);
    bool isL16 = (lane == 16);

    v8f acc0 = {}, acc1 = {}, acc2 = {}, acc3 = {};
    for (int k = k0; k < k1; k += 16) {
        // A operands: 2 consecutive weights per lane per 4-wide K group
        v2f a0 = *(const v2f*)(wrow + k);
        v2f a1 = *(const v2f*)(wrow + k + 4);
        v2f a2 = *(const v2f*)(wrow + k + 8);
        v2f a3 = *(const v2f*)(wrow + k + 12);
        // B operands: x[k..k+15] into column 0 only (uniform scalar loads)
        float4 xa = *(const float4*)(x + k);
        float4 xb = *(const float4*)(x + k + 4);
        float4 xc = *(const float4*)(x + k + 8);
        float4 xd = *(const float4*)(x + k + 12);
        v2f b0, b1, b2, b3;
        b0.x = isL0 ? xa.x : (isL16 ? xa.z : 0.f);
        b0.y = isL0 ? xa.y : (isL16 ? xa.w : 0.f);
        b1.x = isL0 ? xb.x : (isL16 ? xb.z : 0.f);
        b1.y = isL0 ? xb.y : (isL16 ? xb.w : 0.f);
        b2.x = isL0 ? xc.x : (isL16 ? xc.z : 0.f);
        b2.y = isL0 ? xc.y : (isL16 ? xc.w : 0.f);
        b3.x = isL0 ? xd.x : (isL16 ? xd.z : 0.f);
        b3.y = isL0 ? xd.y : (isL16 ? xd.w : 0.f);
        acc0 = __builtin_amdgcn_wmma_f32_16x16x4_f32(false, a0, false, b0, (short)0, acc0, false, false);
        acc1 = __builtin_amdgcn_wmma_f32_16x16x4_f32(false, a1, false, b1, (short)0, acc1, false, false);
        acc2 = __builtin_amdgcn_wmma_f32_16x16x4_f32(false, a2, false, b2, (short)0, acc2, false, false);
        acc3 = __builtin_amdgcn_wmma_f32_16x16x4_f32(false, a3, false, b3, (short)0, acc3, false, false);
    }
    v8f acc = (acc0 + acc1) + (acc2 + acc3);

    if (isL0 || isL16) {
        float* o = pOut + (size_t)blockIdx.y * GATES + row0 + ((lane >> 4) << 3);
        #pragma unroll
        for (int j = 0; j < 8; ++j) o[j] = acc[j];
    }
}

// ---------------------------------------------------------------------------
// Fused: combine k-chunk partials + biases -> gates, then LSTM cell.
// gates split order: i, f, g, o.  1024 threads (grid 4 x 256).
// ---------------------------------------------------------------------------
__global__ void k_lstm_cell(const float* __restrict__ pih, int nih,
                            const float* __restrict__ phh, int nhh,
                            const float* __restrict__ bih, const float* __restrict__ bhh,
                            const float* __restrict__ cprev,
                            float* __restrict__ hout, float* __restrict__ cout) {
    int j = blockIdx.x * blockDim.x + threadIdx.x;   // 0..1023
    float g[4];
    #pragma unroll
    for (int q = 0; q < 4; ++q) {
        int r = q * H_DIM + j;
        float s = bih[r] + bhh[r];
        for (int c = 0; c < nih; ++c) s += pih[(size_t)c * GATES + r];
        for (int c = 0; c < nhh; ++c) s += phh[(size_t)c * GATES + r];
        g[q] = s;
    }
    float ig = 1.f / (1.f + __expf(-g[0]));
    float fg = 1.f / (1.f + __expf(-g[1]));
    float gg = tanhf(g[2]);
    float og = 1.f / (1.f + __expf(-g[3]));
    float c2 = fg * cprev[j] + ig * gg;
    hout[j] = og * tanhf(c2);
    cout[j] = c2;
}

// ---------------------------------------------------------------------------
// Final FC: out[c] = dot(fc_w[c,:], h2) + fc_b[c], C=5 outputs.
// ---------------------------------------------------------------------------
__global__ void k_fc(const float* __restrict__ w, const float* __restrict__ b,
                     const float* __restrict__ h, float* __restrict__ out) {
    __shared__ float red[256];
    int c = blockIdx.x;
    float s = 0.f;
    for (int k = threadIdx.x; k < H_DIM; k += 256) s += w[c * H_DIM + k] * h[k];
    red[threadIdx.x] = s;
    __syncthreads();
    for (int off = 128; off > 0; off >>= 1) {
        if (threadIdx.x < off) red[threadIdx.x] += red[threadIdx.x + off];
        __syncthreads();
    }
    if (threadIdx.x == 0) out[c] = red[0] + b[c];
}

extern "C" void kernel_launch(void* const* d_in, const int* in_sizes, int n_in,
                              void* d_out, int out_size, void* d_ws, size_t ws_size,
                              hipStream_t stream) {
    const float* eh      = (const float*)d_in[0];   // (S, 1, E)
    const float* hidden  = (const float*)d_in[1];   // (2, 1, H)
    const float* cin     = (const float*)d_in[2];   // (2, 1, H)
    // d_in[3] attn_w, d_in[4] attn_b: numerically dead (softmax over size-1 axis)
    const float* w_ih_l0 = (const float*)d_in[5];   // (4H, E)
    const float* w_hh_l0 = (const float*)d_in[6];   // (4H, H)
    const float* b_ih_l0 = (const float*)d_in[7];
    const float* b_hh_l0 = (const float*)d_in[8];
    const float* w_ih_l1 = (const float*)d_in[9];   // (4H, H)
    const float* w_hh_l1 = (const float*)d_in[10];  // (4H, H)
    const float* b_ih_l1 = (const float*)d_in[11];
    const float* b_hh_l1 = (const float*)d_in[12];
    const float* fc_w    = (const float*)d_in[13];  // (5, H)
    const float* fc_b    = (const float*)d_in[14];

    float* ws   = (float*)d_ws;
    float* X0   = ws;                   // 2048  attn_applied
    float* H1   = ws + 2048;            // 1024
    float* C1   = ws + 3072;            // 1024
    float* H2   = ws + 4096;            // 1024
    float* PIH0 = ws + 8192;            // 8 * 4096
    float* PHH0 = PIH0 + 8 * GATES;     // 4 * 4096
    float* PIH1 = PHH0 + 4 * GATES;     // 4 * 4096
    float* PHH1 = PIH1 + 4 * GATES;     // 4 * 4096
    float* PCOL = PHH1 + 4 * GATES;     // NSEG * 2048

    // attention-collapsed column sum (128 MB stream; dominant cost)
    k_col_partial<<<dim3(2, NSEG), 256, 0, stream>>>(eh, PCOL);
    k_col_reduce <<<E_DIM / 256, 256, 0, stream>>>(PCOL, X0);

    // LSTM layer 0: gates = W_ih(4096x2048)@x0 + W_hh(4096x1024)@h0 + biases
    k_gemv_wmma<<<dim3(GATES / 16, 8), 32, 0, stream>>>(w_ih_l0, X0,     E_DIM, 256, PIH0);
    k_gemv_wmma<<<dim3(GATES / 16, 4), 32, 0, stream>>>(w_hh_l0, hidden, H_DIM, 256, PHH0);
    k_lstm_cell<<<4, 256, 0, stream>>>(PIH0, 8, PHH0, 4, b_ih_l0, b_hh_l0, cin, H1, C1);

    // LSTM layer 1 (c2 of layer 1 is unused downstream; C1 reused as scratch)
    k_gemv_wmma<<<dim3(GATES / 16, 4), 32, 0, stream>>>(w_ih_l1, H1,             H_DIM, 256, PIH1);
    k_gemv_wmma<<<dim3(GATES / 16, 4), 32, 0, stream>>>(w_hh_l1, hidden + H_DIM, H_DIM, 256, PHH1);
    k_lstm_cell<<<4, 256, 0, stream>>>(PIH1, 4, PHH1, 4, b_ih_l1, b_hh_l1, cin + H_DIM, H2, C1);

    // final projection -> d_out (5 floats)
    k_fc<<<5, 256, 0, stream>>>(fc_w, fc_b, H2, (float*)d_out);
}